// SinkhornLoss_23261542875721
// MI455X (gfx1250) — compile-verified
//
#include <hip/hip_runtime.h>

// Sinkhorn loss, b=16, n=48*48=2304, eps=0.01, 5 iterations.
// Trans-bound: C recomputed on the fly from a pre-scaled LDS column table,
// LSE row sums reduced with per-lane partials + v_wmma_f32_16x16x4_f32.

typedef __attribute__((ext_vector_type(2))) float v2f;
typedef __attribute__((ext_vector_type(8))) float v8f;

namespace {
constexpr int kN  = 2304;          // 48*48 grid points
constexpr int kB  = 16;            // batch
constexpr int kBN = kN * kB;
constexpr float kEps   = 0.01f;
constexpr float kSc    = 10.0f / 48.0f;  // grid scale * sqrt(1/eps): (10*d)^2 = d^2/eps
constexpr int kTilesX = kN / 64;   // 36 blocks of 64 rows (4 waves x 16 rows)
constexpr int kTab    = kN / 2;    // 1152 float4 column-table entries
}

// ---------------------------------------------------------------------------
// Init: log_mu / log_nu (normalized clipped mass), zero potentials.
// grid = (16 batches, 2 arrays), block = 256
// ---------------------------------------------------------------------------
__global__ __launch_bounds__(256) void sk_init(const float* __restrict__ y,
                                               const float* __restrict__ yt,
                                               float* __restrict__ log_mu,
                                               float* __restrict__ log_nu,
                                               float* __restrict__ u,
                                               float* __restrict__ v) {
  const int b = blockIdx.x;
  const int which = blockIdx.y;
  const float* src = (which ? yt : y) + b * kN;
  float* dst = (which ? log_nu : log_mu) + b * kN;
  float* pot = (which ? v : u) + b * kN;
  const int tid = threadIdx.x;
  __shared__ float red[256];

  float s = 0.0f;
  for (int t = tid; t < kN; t += 256) {
    s += fminf(fmaxf(src[t], 0.0f), 1e9f) + 1e-9f;
  }
  red[tid] = s;
  __syncthreads();
  for (int o = 128; o > 0; o >>= 1) {
    if (tid < o) red[tid] += red[tid + o];
    __syncthreads();
  }
  const float ltot = __logf(red[0]);
  for (int t = tid; t < kN; t += 256) {
    const float m = fminf(fmaxf(src[t], 0.0f), 1e9f) + 1e-9f;
    dst[t] = __logf(m) - ltot;   // log(mass/total)
    pot[t] = 0.0f;
  }
}

// Build the per-column table: {100*v_j, 100*v_{j+1}, 10*gx_j, 10*gy_j} for even j.
__device__ __forceinline__ void fill_tab(float4* stab, const float* pcb, int tid) {
  for (int e = tid; e < kTab; e += 128) {     // 1152/128 = 9, uniform trip count
    const int j  = e << 1;
    const int jy = j / 48;
    const int jx = j - jy * 48;
    float4 t;
    t.x = pcb[j]     * 100.0f;
    t.y = pcb[j + 1] * 100.0f;
    t.z = (float)jx * kSc;
    t.w = (float)jy * kSc;
    stab[e] = t;
  }
}

// ---------------------------------------------------------------------------
// One LSE half-iteration:
//   pr_i += eps*(logm_i - log(sum_j exp((pr_i + pc_j - C_ij)/eps) + 1e-6))
// C symmetric -> u-update and v-update share this kernel with swapped args.
// grid = (36 row-tiles, 16 batches), block = 128 (4 waves, 16 rows each).
// Lane -> WMMA 16x4 f32 A layout: M = lane&15, K = 2*(lane>>4) + {0,1}.
// ---------------------------------------------------------------------------
__global__ __launch_bounds__(128) void sk_lse_update(
    const float* __restrict__ logm,   // [B*N] log_mu or log_nu
    float* __restrict__ pr,           // [B*N] potential being updated
    const float* __restrict__ pc) {   // [B*N] other potential (read-only)
  __shared__ float4 stab[kTab];
  const int b = blockIdx.y;
  const int tid = threadIdx.x;
  fill_tab(stab, pc + b * kN, tid);
  __syncthreads();

  const int wave = tid >> 5;
  const int lane = tid & 31;
  const int m    = lane & 15;
  const int kb   = (lane >> 4) << 1;        // 0 or 2
  const int rbase = blockIdx.x * 64 + wave * 16;
  const int row   = rbase + m;
  const int iy = row / 48;
  const int ix = row - iy * 48;
  const float hyi = (float)iy * kSc;        // 10*gy_i
  const float hxi = (float)ix * kSc;        // 10*gx_i
  const float u100 = pr[b * kN + row] * 100.0f;

  v2f pacc;
  pacc.x = 0.0f;
  pacc.y = 0.0f;

  for (int e = kb >> 1; e < kTab; e += 2) {  // this lane's column pairs
    const float4 t = stab[e];
    const float dy  = hyi - t.w;
    const float dy2 = dy * dy;
    const float dx0 = hxi - t.z;
    const float dx1 = dx0 - kSc;
    const float a0 = (u100 + t.x) - fmaf(dx0, dx0, dy2);  // (u+v-C)/eps
    const float a1 = (u100 + t.y) - fmaf(dx1, dx1, dy2);
    pacc.x += __expf(a0);
    pacc.y += __expf(a1);
  }

  // Cross-lane/K reduction: D = pacc(16x4) x ones(4x16) -> row sums in D.
  v2f ones;
  ones.x = 1.0f;
  ones.y = 1.0f;
  v8f acc = {};
  acc = __builtin_amdgcn_wmma_f32_16x16x4_f32(
      /*neg_a=*/false, pacc, /*neg_b=*/false, ones,
      /*c_mod=*/(short)0, acc, /*reuse_a=*/false, /*reuse_b=*/false);

  // D layout: VGPR r, lanes 0-15 -> M=r ; lanes 16-31 -> M=8+r (all N equal).
  if (lane == 0 || lane == 16) {
    const int roff = (lane == 16) ? 8 : 0;
#pragma unroll
    for (int r = 0; r < 8; ++r) {
      const int i = rbase + roff + r;
      const float lse = __logf(acc[r] + 1e-6f);       // non-shifted, as reference
      const float lm  = logm[b * kN + i];
      pr[b * kN + i] = kEps * (lm - lse) + pr[b * kN + i];
    }
  }
}

// ---------------------------------------------------------------------------
// Final contraction: per-block partial of sum_ij exp(M_ij) * C_ij.
// Same tile structure; WMMA row-sum reduction, then tiny LDS combine.
// ---------------------------------------------------------------------------
__global__ __launch_bounds__(128) void sk_cost(const float* __restrict__ u,
                                               const float* __restrict__ v,
                                               float* __restrict__ partials) {
  __shared__ float4 stab[kTab];
  __shared__ float red[8];                  // 4 waves x 2 half-wave results
  const int b = blockIdx.y;
  const int tid = threadIdx.x;
  fill_tab(stab, v + b * kN, tid);
  __syncthreads();

  const int wave = tid >> 5;
  const int lane = tid & 31;
  const int m    = lane & 15;
  const int kb   = (lane >> 4) << 1;
  const int row  = blockIdx.x * 64 + wave * 16 + m;
  const int iy = row / 48;
  const int ix = row - iy * 48;
  const float hyi = (float)iy * kSc;
  const float hxi = (float)ix * kSc;
  const float u100 = u[b * kN + row] * 100.0f;

  v2f pacc;                                  // accumulates exp(arg) * (100*C)
  pacc.x = 0.0f;
  pacc.y = 0.0f;

  for (int e = kb >> 1; e < kTab; e += 2) {
    const float4 t = stab[e];
    const float dy  = hyi - t.w;
    const float dy2 = dy * dy;
    const float dx0 = hxi - t.z;
    const float dx1 = dx0 - kSc;
    const float c0 = fmaf(dx0, dx0, dy2);    // C_ij / eps
    const float c1 = fmaf(dx1, dx1, dy2);
    pacc.x += __expf((u100 + t.x) - c0) * c0;
    pacc.y += __expf((u100 + t.y) - c1) * c1;
  }

  v2f ones;
  ones.x = 1.0f;
  ones.y = 1.0f;
  v8f acc = {};
  acc = __builtin_amdgcn_wmma_f32_16x16x4_f32(
      false, pacc, false, ones, (short)0, acc, false, false);

  if (lane == 0 || lane == 16) {
    float s = 0.0f;
#pragma unroll
    for (int r = 0; r < 8; ++r) s += acc[r];
    red[wave * 2 + (lane == 16 ? 1 : 0)] = s;
  }
  __syncthreads();
  if (tid == 0) {
    float s = 0.0f;
#pragma unroll
    for (int r = 0; r < 8; ++r) s += red[r];
    partials[b * kTilesX + blockIdx.x] = s * kEps;   // undo the /eps scaling of C
  }
}

// ---------------------------------------------------------------------------
// Reduce the 576 partials, mean over batch -> d_out[0].
// ---------------------------------------------------------------------------
__global__ __launch_bounds__(256) void sk_final(const float* __restrict__ partials,
                                                float* __restrict__ out) {
  __shared__ float red[256];
  const int tid = threadIdx.x;
  float s = 0.0f;
  for (int t = tid; t < kB * kTilesX; t += 256) s += partials[t];
  red[tid] = s;
  __syncthreads();
  for (int o = 128; o > 0; o >>= 1) {
    if (tid < o) red[tid] += red[tid + o];
    __syncthreads();
  }
  if (tid == 0) out[0] = red[0] * (1.0f / (float)kB);
}

// ---------------------------------------------------------------------------
extern "C" void kernel_launch(void* const* d_in, const int* in_sizes, int n_in,
                              void* d_out, int out_size, void* d_ws, size_t ws_size,
                              hipStream_t stream) {
  const float* y  = (const float*)d_in[0];
  const float* yt = (const float*)d_in[1];
  float* ws = (float*)d_ws;
  float* log_mu   = ws;                 // [B*N]
  float* log_nu   = ws + kBN;           // [B*N]
  float* u        = ws + 2 * kBN;       // [B*N]
  float* v        = ws + 3 * kBN;       // [B*N]
  float* partials = ws + 4 * kBN;       // [576]

  sk_init<<<dim3(kB, 2), 256, 0, stream>>>(y, yt, log_mu, log_nu, u, v);
  for (int it = 0; it < 5; ++it) {
    sk_lse_update<<<dim3(kTilesX, kB), 128, 0, stream>>>(log_mu, u, v); // u-update
    sk_lse_update<<<dim3(kTilesX, kB), 128, 0, stream>>>(log_nu, v, u); // v-update (C symmetric)
  }
  sk_cost<<<dim3(kTilesX, kB), 128, 0, stream>>>(u, v, partials);
  sk_final<<<1, 256, 0, stream>>>(partials, (float*)d_out);
}